// CGCNN_42442866819218
// MI455X (gfx1250) — compile-verified
//
#include <hip/hip_runtime.h>
#include <math.h>

#define N_NODES  50000
#define N_EDGES  800000
#define ORIG_FEA 92
#define NBR_FEA  41
#define HID      64
#define N_CONV   3
#define BN_EPS   1e-5f

typedef __attribute__((ext_vector_type(2))) float v2f;
typedef __attribute__((ext_vector_type(8))) float v8f;

// D = A(16x4 f32) * B(4x16 f32) + C(16x16 f32), wave32 WMMA
__device__ __forceinline__ v8f wmma_f32(v2f a, v2f b, v8f c) {
  return __builtin_amdgcn_wmma_f32_16x16x4_f32(false, a, false, b, (short)0, c,
                                               false, false);
}

__device__ __forceinline__ float sigmoid_f(float x) {
  return __builtin_amdgcn_rcpf(1.f + __expf(-x));
}
__device__ __forceinline__ float softplus_f(float x) {
  return (x > 20.f) ? x : __logf(1.f + __expf(x));
}

// ---------------------------------------------------------------------------
// C[M,64] = A[M,K] @ B[K,64] + bias  (optionally duplicated into Cdup).
// B is staged TRANSPOSED in LDS (Bt[col*SK + k]) so each WMMA B-fragment is a
// single contiguous ds_load_b64 straight into an even VGPR pair (no movs).
// M must be a multiple of 16; K%4 tail via clamped loads + lane selects.
// ---------------------------------------------------------------------------
template <int K>
__global__ void gemm_embed(const float* __restrict__ A, const float* __restrict__ B,
                           const float* __restrict__ bias, float* __restrict__ C,
                           float* __restrict__ Cdup, int M) {
  constexpr int SK = (K & 1) ? (K + 3) : (K + 2);   // even row stride (floats)
  __shared__ float ldsBt[64 * SK];

  // stage B transposed: coalesced float4 global reads, scattered b32 LDS writes
  for (int c = threadIdx.x; c < K * 16; c += 256) {
    const int k = c >> 4, c4 = (c & 15) << 2;
    const float4 w = *(const float4*)(B + (size_t)k * HID + c4);
    ldsBt[(c4 + 0) * SK + k] = w.x;
    ldsBt[(c4 + 1) * SK + k] = w.y;
    ldsBt[(c4 + 2) * SK + k] = w.z;
    ldsBt[(c4 + 3) * SK + k] = w.w;
  }
  __syncthreads();

  const int lane = threadIdx.x & 31;
  const int wave = threadIdx.x >> 5;
  const int mt   = blockIdx.x * (blockDim.x >> 5) + wave;
  if (mt * 16 >= M) return;                         // wave-uniform exit
  const int half = lane >> 4;
  const int l16  = lane & 15;
  const float* __restrict__ Arow = A + (size_t)(mt * 16 + l16) * K;

  v8f acc[4] = {};
#pragma unroll
  for (int kk = 0; kk < K / 4; ++kk) {
    const int k0 = kk * 4 + half * 2;
    v2f a;
    if constexpr ((K & 1) == 0) {
      a = *(const v2f*)(Arow + k0);                 // 8B-aligned when K even
    } else {
      a.x = Arow[k0];
      a.y = Arow[k0 + 1];
    }
#pragma unroll
    for (int nt = 0; nt < 4; ++nt) {
      const v2f b = *(const v2f*)(&ldsBt[(nt * 16 + l16) * SK + k0]);
      acc[nt] = wmma_f32(a, b, acc[nt]);
    }
  }
  if constexpr ((K & 3) != 0) {
    // tail k-quad: unconditional in-bounds loads, zero invalid lanes by select
    const int k0  = (K & ~3) + half * 2;
    const int k0c = (k0 < K)     ? k0     : K - 1;
    const int k1c = (k0 + 1 < K) ? k0 + 1 : K - 1;
    const bool vld0 = k0 < K, vld1 = (k0 + 1) < K;
    const float a0 = Arow[k0c], a1 = Arow[k1c];
    v2f a;
    a.x = vld0 ? a0 : 0.f;
    a.y = vld1 ? a1 : 0.f;
#pragma unroll
    for (int nt = 0; nt < 4; ++nt) {
      const int col = nt * 16 + l16;
      const float b0 = ldsBt[col * SK + k0c];
      const float b1 = ldsBt[col * SK + k1c];
      v2f b;
      b.x = vld0 ? b0 : 0.f;
      b.y = vld1 ? b1 : 0.f;
      acc[nt] = wmma_f32(a, b, acc[nt]);
    }
  }
#pragma unroll
  for (int nt = 0; nt < 4; ++nt) {
    const int col = nt * 16 + l16;
    const float bv = bias[col];
#pragma unroll
    for (int j = 0; j < 8; ++j) {
      const int r = mt * 16 + j + half * 8;   // C/D layout: VGPR j -> M=j / M=j+8
      const float v = acc[nt][j] + bv;
      C[(size_t)r * HID + col] = v;
      if (Cdup) Cdup[(size_t)r * HID + col] = v;
    }
  }
}

// ---------------------------------------------------------------------------
// Dual node transform: Cf = A @ Bf, Cs = A @ Bs  (K = 64, no bias).
// A fragments loaded once, reused for both weight matrices.
// ---------------------------------------------------------------------------
__global__ void gemm_dual64(const float* __restrict__ A,
                            const float* __restrict__ Bf, const float* __restrict__ Bs,
                            float* __restrict__ Cf, float* __restrict__ Cs, int M) {
  constexpr int SK = 66;
  __shared__ float ldsBt[2][64 * SK];
  for (int c = threadIdx.x; c < 64 * 16; c += 256) {
    const int k = c >> 4, c4 = (c & 15) << 2;
    const float4 wf = *(const float4*)(Bf + (size_t)k * HID + c4);
    const float4 ws = *(const float4*)(Bs + (size_t)k * HID + c4);
    ldsBt[0][(c4 + 0) * SK + k] = wf.x;
    ldsBt[0][(c4 + 1) * SK + k] = wf.y;
    ldsBt[0][(c4 + 2) * SK + k] = wf.z;
    ldsBt[0][(c4 + 3) * SK + k] = wf.w;
    ldsBt[1][(c4 + 0) * SK + k] = ws.x;
    ldsBt[1][(c4 + 1) * SK + k] = ws.y;
    ldsBt[1][(c4 + 2) * SK + k] = ws.z;
    ldsBt[1][(c4 + 3) * SK + k] = ws.w;
  }
  __syncthreads();

  const int lane = threadIdx.x & 31;
  const int wave = threadIdx.x >> 5;
  const int mt   = blockIdx.x * (blockDim.x >> 5) + wave;
  if (mt * 16 >= M) return;
  const int half = lane >> 4;
  const int l16  = lane & 15;
  const float* __restrict__ Arow = A + (size_t)(mt * 16 + l16) * HID;

  v8f accF[4] = {};
  v8f accS[4] = {};
#pragma unroll
  for (int kk = 0; kk < 16; ++kk) {
    const int k0 = kk * 4 + half * 2;
    const v2f a = *(const v2f*)(Arow + k0);
#pragma unroll
    for (int nt = 0; nt < 4; ++nt) {
      const v2f b = *(const v2f*)(&ldsBt[0][(nt * 16 + l16) * SK + k0]);
      accF[nt] = wmma_f32(a, b, accF[nt]);
    }
#pragma unroll
    for (int nt = 0; nt < 4; ++nt) {
      const v2f b = *(const v2f*)(&ldsBt[1][(nt * 16 + l16) * SK + k0]);
      accS[nt] = wmma_f32(a, b, accS[nt]);
    }
  }
#pragma unroll
  for (int nt = 0; nt < 4; ++nt) {
    const int col = nt * 16 + l16;
#pragma unroll
    for (int j = 0; j < 8; ++j) {
      const int r = mt * 16 + j + half * 8;
      Cf[(size_t)r * HID + col] = accF[nt][j];
      Cs[(size_t)r * HID + col] = accS[nt][j];
    }
  }
}

// ---------------------------------------------------------------------------
// Fused CGConv edge kernel. Per wave: one tile of 16 edges.
//   preF = e_tile @ WfE  (WMMA; weights staged TRANSPOSED in LDS)
//   gate = sigmoid(preF + TtF[tgt] + TsF[src] + bf)
//   core = softplus(preS + TtS[tgt] + TsS[src] + bs)
//   atomicAdd(hOut[tgt], gate*core)   (hOut pre-initialized with h -> residual)
// ---------------------------------------------------------------------------
#define LP 68   // e-tile LDS row stride (conflict-free b64 A-frag reads)
#define WS 66   // transposed weight LDS row stride
__global__ void cg_edge_fused(const float* __restrict__ Ee,      // [E,64]
                              const int*   __restrict__ srcIdx,
                              const int*   __restrict__ tgtIdx,
                              const float* __restrict__ WfE,     // [64,64]
                              const float* __restrict__ WsE,     // [64,64]
                              const float* __restrict__ bf,
                              const float* __restrict__ bs,
                              const float* __restrict__ TtF,
                              const float* __restrict__ TsF,
                              const float* __restrict__ TtS,
                              const float* __restrict__ TsS,
                              float* __restrict__ hOut,
                              int nTiles) {
  __shared__ float ldsWt[2][64 * WS];  // 33792 B (transposed: [col][k])
  __shared__ float ldsE[4][16 * LP];   // 17408 B
  const int t = threadIdx.x;           // 128 threads = 4 waves

  // stage both 64x64 weight blocks transposed
  for (int c = t; c < 64 * 16; c += 128) {
    const int k = c >> 4, c4 = (c & 15) << 2;
    const float4 wf = *(const float4*)(WfE + (size_t)k * HID + c4);
    const float4 ws = *(const float4*)(WsE + (size_t)k * HID + c4);
    ldsWt[0][(c4 + 0) * WS + k] = wf.x;
    ldsWt[0][(c4 + 1) * WS + k] = wf.y;
    ldsWt[0][(c4 + 2) * WS + k] = wf.z;
    ldsWt[0][(c4 + 3) * WS + k] = wf.w;
    ldsWt[1][(c4 + 0) * WS + k] = ws.x;
    ldsWt[1][(c4 + 1) * WS + k] = ws.y;
    ldsWt[1][(c4 + 2) * WS + k] = ws.z;
    ldsWt[1][(c4 + 3) * WS + k] = ws.w;
  }
  __syncthreads();

  const int lane = t & 31, wave = t >> 5;
  const int half = lane >> 4, l16 = lane & 15;
  const int tile = blockIdx.x * 4 + wave;
  if (tile >= nTiles) return;                // wave-uniform
  const int eb = tile * 16;
  float* myE = ldsE[wave];

  // stage this wave's 16x64 e-tile (coalesced b128 loads)
#pragma unroll
  for (int c = lane; c < 16 * 16; c += 32) {
    const int r = c >> 4, c4 = (c & 15) << 2;
    *(float4*)(&myE[r * LP + c4]) = *(const float4*)(Ee + (size_t)(eb + r) * HID + c4);
  }

  v8f aF[4] = {};
  v8f aS[4] = {};
#pragma unroll
  for (int kk = 0; kk < 16; ++kk) {
    const int k0 = kk * 4 + half * 2;
    const v2f a = *(const v2f*)(&myE[l16 * LP + k0]);
#pragma unroll
    for (int nt = 0; nt < 4; ++nt) {
      const v2f b = *(const v2f*)(&ldsWt[0][(nt * 16 + l16) * WS + k0]);
      aF[nt] = wmma_f32(a, b, aF[nt]);
    }
#pragma unroll
    for (int nt = 0; nt < 4; ++nt) {
      const v2f b = *(const v2f*)(&ldsWt[1][(nt * 16 + l16) * WS + k0]);
      aS[nt] = wmma_f32(a, b, aS[nt]);
    }
  }

  float bfv[4], bsv[4];
#pragma unroll
  for (int nt = 0; nt < 4; ++nt) {
    bfv[nt] = bf[nt * 16 + l16];
    bsv[nt] = bs[nt * 16 + l16];
  }
#pragma unroll
  for (int j = 0; j < 8; ++j) {
    const int eIdx = eb + j + half * 8;       // C/D layout: VGPR j -> M=j / M=j+8
    const int tb = tgtIdx[eIdx] * HID;
    const int sb = srcIdx[eIdx] * HID;
#pragma unroll
    for (int nt = 0; nt < 4; ++nt) {
      const int col = nt * 16 + l16;
      const float g = aF[nt][j] + TtF[tb + col] + TsF[sb + col] + bfv[nt];
      const float s = aS[nt][j] + TtS[tb + col] + TsS[sb + col] + bsv[nt];
      atomicAdd(hOut + tb + col, sigmoid_f(g) * softplus_f(s));
    }
  }
}

// ---------------------------------------------------------------------------
// BatchNorm (training-mode batch stats, biased variance) + optional ReLU
// ---------------------------------------------------------------------------
__global__ void zero128(float* p) { p[threadIdx.x] = 0.f; }

__global__ void bn_stats(const float* __restrict__ X, float* __restrict__ stats,
                         int nRows) {
  __shared__ float red[256];
  const int t = threadIdx.x;
  const int col = t & 63;
  const int grp = t >> 6;
  float s = 0.f, q = 0.f;
  for (int r = blockIdx.x * 4 + grp; r < nRows; r += gridDim.x * 4) {
    const float v = X[(size_t)r * HID + col];
    s += v; q += v * v;
  }
  red[t] = s;
  __syncthreads();
  float ts = 0.f;
  if (grp == 0) ts = red[col] + red[col + 64] + red[col + 128] + red[col + 192];
  __syncthreads();
  red[t] = q;
  __syncthreads();
  if (grp == 0) {
    const float tq = red[col] + red[col + 64] + red[col + 128] + red[col + 192];
    atomicAdd(&stats[col], ts);
    atomicAdd(&stats[64 + col], tq);
  }
}

// Y (and optionally Y2) = BN(X) [+ReLU]; Y2 pre-seeds next layer's residual acc
__global__ void bn_apply(const float* __restrict__ X, const float* __restrict__ stats,
                         const float* __restrict__ gamma, const float* __restrict__ beta,
                         float* __restrict__ Y, float* __restrict__ Y2,
                         int nRows, int relu) {
  const int idx = blockIdx.x * blockDim.x + threadIdx.x;
  if (idx >= nRows * HID) return;
  const int col = idx & 63;
  const float invN = 1.f / (float)nRows;
  const float mean = stats[col] * invN;
  const float var  = stats[64 + col] * invN - mean * mean;
  const float rstd = rsqrtf(var + BN_EPS);
  float v = (X[idx] - mean) * rstd * gamma[col] + beta[col];
  if (relu) v = fmaxf(v, 0.f);
  Y[idx] = v;
  if (Y2) Y2[idx] = v;
}

// ---------------------------------------------------------------------------
extern "C" void kernel_launch(void* const* d_in, const int* in_sizes, int n_in,
                              void* d_out, int out_size, void* d_ws, size_t ws_size,
                              hipStream_t stream) {
  (void)in_sizes; (void)n_in; (void)out_size; (void)ws_size;
  const float* x     = (const float*)d_in[0];
  const float* ea    = (const float*)d_in[1];
  const int*   ei    = (const int*)  d_in[2];
  const float* W1    = (const float*)d_in[3];
  const float* b1    = (const float*)d_in[4];
  const float* W2    = (const float*)d_in[5];
  const float* b2    = (const float*)d_in[6];
  const float* Wf    = (const float*)d_in[7];
  const float* bf    = (const float*)d_in[8];
  const float* Ws    = (const float*)d_in[9];
  const float* bs    = (const float*)d_in[10];
  const float* gamma = (const float*)d_in[11];
  const float* beta  = (const float*)d_in[12];
  const int* srcI = ei;
  const int* tgtI = ei + N_EDGES;

  const size_t N64 = (size_t)N_NODES * HID;
  float* e   = (float*)d_ws;                    // [E,64]   204.8 MB
  float* hA  = e   + (size_t)N_EDGES * HID;     // [N,64]  current features
  float* hB  = hA  + N64;                       // [N,64]  residual accumulator
  float* TtF = hB  + N64;                       // [N,64] x4 node transforms
  float* TsF = TtF + N64;
  float* TtS = TsF + N64;
  float* TsS = TtS + N64;
  float* stats = TsS + N64;                     // 128 floats

  const int nodeBlocks = (N_NODES / 16 + 7) / 8;            // 391
  const int edgeTiles  = N_EDGES / 16;                      // 50000 (exact)

  // embeddings (node embed writes hA and seeds residual hB)
  gemm_embed<ORIG_FEA><<<nodeBlocks, 256, 0, stream>>>(x, W1, b1, hA, hB, N_NODES);
  gemm_embed<NBR_FEA><<<edgeTiles / 8, 256, 0, stream>>>(ea, W2, b2, e, nullptr, N_EDGES);

  for (int i = 0; i < N_CONV; ++i) {
    const float* Wfi = Wf + (size_t)i * 3 * HID * HID;
    const float* Wsi = Ws + (size_t)i * 3 * HID * HID;
    // node-side transforms (gate: target/source blocks; core: target/source blocks)
    gemm_dual64<<<nodeBlocks, 256, 0, stream>>>(hA, Wfi, Wfi + 64 * HID, TtF, TsF, N_NODES);
    gemm_dual64<<<nodeBlocks, 256, 0, stream>>>(hA, Wsi, Wsi + 64 * HID, TtS, TsS, N_NODES);
    // fused e@W_e (WMMA) + gather + sigmoid*softplus + scatter-add into hB
    cg_edge_fused<<<edgeTiles / 4, 128, 0, stream>>>(
        e, srcI, tgtI, Wfi + 128 * HID, Wsi + 128 * HID,
        bf + i * HID, bs + i * HID, TtF, TsF, TtS, TsS, hB, edgeTiles);
    // batchnorm (+relu except last layer); also re-seed hB for next layer
    zero128<<<1, 128, 0, stream>>>(stats);
    bn_stats<<<512, 256, 0, stream>>>(hB, stats, N_NODES);
    const bool last = (i == N_CONV - 1);
    bn_apply<<<(int)(N64 + 255) / 256, 256, 0, stream>>>(
        hB, stats, gamma + i * HID, beta + i * HID,
        last ? (float*)d_out : hA, last ? nullptr : hB, N_NODES, last ? 0 : 1);
  }
}